// LIF_layer_17497696764281
// MI455X (gfx1250) — compile-verified
//
#include <hip/hip_runtime.h>
#include <cstdint>

// Use a raw ext_vector so __builtin_nontemporal_load/store accept it directly.
typedef float float4v __attribute__((ext_vector_type(4)));

#define TPB    256                 // 8 waves (wave32) per block
#define UNROLL 4                   // float4 slots per thread per tile
#define TILE4  (TPB * UNROLL)      // 1024 float4 = 16 KB per tile
#define MAXGRID 2048

// ---------------------------------------------------------------------------
// Streaming LIF spike kernel: out = (in >= 1.0f) ? 1.0f : 0.0f
// Double-buffered async global->LDS pipeline (CDNA5 ASYNCcnt path):
//   issue(tile i+1)  ||  wait(asynccnt<=UNROLL)  ->  process(tile i)
// Each lane consumes exactly the LDS bytes it requested, so no barrier is
// required — only the per-wave async counter wait. Both 256 MiB streams are
// read/written once, so loads and stores carry non-temporal hints to avoid
// L2 (192 MB) thrash.
// ---------------------------------------------------------------------------
__global__ __launch_bounds__(TPB)
void lif_spike_async_kernel(const float* __restrict__ in,
                            float* __restrict__ out,
                            long long nTiles) {
    __shared__ float4v buf[2][TILE4];     // 32 KB double buffer

    const int tid = threadIdx.x;
    const float4v* __restrict__ in4  = (const float4v*)in;
    float4v* __restrict__       out4 = (float4v*)out;

    long long tile = blockIdx.x;
    if (tile >= nTiles) return;
    const long long tileStride = gridDim.x;

    // Wave-relative LDS byte addresses for this thread's slots.
    // (generic -> flat truncation: low 32 bits of a __shared__ flat address
    //  are the addrspace(3) offset the async engine expects in VDST)
    uint32_t ldsOff[2][UNROLL];
#pragma unroll
    for (int b = 0; b < 2; ++b)
#pragma unroll
        for (int j = 0; j < UNROLL; ++j)
            ldsOff[b][j] = (uint32_t)(uintptr_t)&buf[b][j * TPB + tid];

    auto issue = [&](int b, long long base4) {
#pragma unroll
        for (int j = 0; j < UNROLL; ++j) {
            unsigned long long ga =
                (unsigned long long)(in4 + base4 + (long long)(j * TPB + tid));
            // CDNA5 async copy: HBM -> LDS, no VGPR destination, ASYNCcnt++.
            // Non-temporal: the 256 MiB input stream is read exactly once.
            asm volatile("global_load_async_to_lds_b128 %0, %1, off th:TH_LOAD_NT"
                         :: "v"(ldsOff[b][j]), "v"(ga)
                         : "memory");
        }
    };

    auto process = [&](int b, long long base4) {
#pragma unroll
        for (int j = 0; j < UNROLL; ++j) {
            float4v v = buf[b][j * TPB + tid];      // ds_load_b128
            float4v r;
            r.x = (v.x >= 1.0f) ? 1.0f : 0.0f;
            r.y = (v.y >= 1.0f) ? 1.0f : 0.0f;
            r.z = (v.z >= 1.0f) ? 1.0f : 0.0f;
            r.w = (v.w >= 1.0f) ? 1.0f : 0.0f;
            // Streaming output, zero reuse: non-temporal b128 store keeps the
            // 256 MiB result stream out of the 192 MB L2.
            __builtin_nontemporal_store(r, out4 + base4 + (long long)(j * TPB + tid));
        }
    };

    int b = 0;
    long long cur = tile;
    issue(b, cur * TILE4);                           // prime the pipeline
    for (long long nxt = cur + tileStride; nxt < nTiles; nxt += tileStride) {
        issue(b ^ 1, nxt * TILE4);                   // prefetch next tile
        // First UNROLL async ops (current tile) complete when outstanding<=UNROLL
        asm volatile("s_wait_asynccnt 4" ::: "memory");
        process(b, cur * TILE4);
        b ^= 1;
        cur = nxt;
    }
    asm volatile("s_wait_asynccnt 0" ::: "memory");  // drain
    process(b, cur * TILE4);
}

// Scalar tail for any elements not covered by whole tiles (empty for the
// reference shape 16*32*128*32*32, but kept for generality/determinism).
__global__ __launch_bounds__(TPB)
void lif_spike_tail_kernel(const float* __restrict__ in,
                           float* __restrict__ out,
                           long long start, long long n) {
    long long i = start + (long long)blockIdx.x * blockDim.x + threadIdx.x;
    if (i < n) out[i] = (in[i] >= 1.0f) ? 1.0f : 0.0f;
}

extern "C" void kernel_launch(void* const* d_in, const int* in_sizes, int n_in,
                              void* d_out, int out_size, void* d_ws, size_t ws_size,
                              hipStream_t stream) {
    (void)n_in; (void)d_ws; (void)ws_size;
    const float* in  = (const float*)d_in[0];
    float*       out = (float*)d_out;

    const long long n      = (long long)in_sizes[0];   // == out_size
    const long long n4     = n / 4;
    const long long nTiles = n4 / TILE4;

    if (nTiles > 0) {
        const int grid = (int)(nTiles < MAXGRID ? nTiles : MAXGRID);
        hipLaunchKernelGGL(lif_spike_async_kernel, dim3(grid), dim3(TPB), 0, stream,
                           in, out, nTiles);
    }

    const long long handled = nTiles * (long long)TILE4 * 4;
    const long long tail    = n - handled;
    if (tail > 0) {
        const int tgrid = (int)((tail + TPB - 1) / TPB);
        hipLaunchKernelGGL(lif_spike_tail_kernel, dim3(tgrid), dim3(TPB), 0, stream,
                           in, out, handled, n);
    }
}